// SkeletonPreprocessor_46797963657351
// MI455X (gfx1250) — compile-verified
//
#include <hip/hip_runtime.h>
#include <cfloat>
#include <cstdint>

// SkeletonPreprocessor: per item (17 joints x 2 coords) min/max-normalize.
// Memory-bound: 570MB total traffic -> ~24.5us floor at 23.3 TB/s.
// Data path: async global->LDS stage-in and async LDS->global stage-out
// (both ASYNCcnt-tracked, saddr-form), conflict-free LDS compute between.

#define THREADS          256
#define ITEMS_PER_BLOCK  256
#define JOINTS           17
#define FLOATS_PER_ITEM  (JOINTS * 2)                              // 34
#define VEC_PER_BLOCK    (ITEMS_PER_BLOCK * FLOATS_PER_ITEM / 4)   // 2176
#define STAGE_ITERS      ((VEC_PER_BLOCK + THREADS - 1) / THREADS) // 9

__global__ __launch_bounds__(THREADS)
void SkeletonPreprocessor_46797963657351_kernel(const float* __restrict__ in,
                                                float* __restrict__ out,
                                                int n_items)
{
    // Dynamic LDS only (no static __shared__) => smem[0] is LDS byte offset 0,
    // so the integer LDS addresses fed to the async-copy asm match smem[].
    extern __shared__ float smem[];

    const int tid = threadIdx.x;
    const long long blockItemBase  = (long long)blockIdx.x * ITEMS_PER_BLOCK;
    const long long blockFloatBase = blockItemBase * FLOATS_PER_ITEM;
    const long long totalFloats    = (long long)n_items * FLOATS_PER_ITEM;

    // Uniform per-block base pointers -> SGPRs; per-lane offsets are 32-bit.
    const float* __restrict__ inBase  = in  + blockFloatBase;
    float*       __restrict__ outBase = out + blockFloatBase;
    const bool fullBlock =
        (blockFloatBase + (long long)VEC_PER_BLOCK * 4 <= totalFloats);

    // ---------- Stage 1: coalesced async copy global -> LDS (ASYNCcnt)
    if (fullBlock) {
        const uint64_t sbase = (uint64_t)(uintptr_t)inBase;
        #pragma unroll
        for (int i = 0; i < STAGE_ITERS; ++i) {
            const int v = tid + i * THREADS;
            if (i * THREADS + THREADS <= VEC_PER_BLOCK || v < VEC_PER_BLOCK) {
                const unsigned lds_off = (unsigned)(v * 16);   // LDS byte address
                const unsigned goff    = (unsigned)(v * 16);   // 32-bit voffset
                asm volatile("global_load_async_to_lds_b128 %0, %1, %2"
                             :: "v"(lds_off), "v"(goff), "s"(sbase)
                             : "memory");
            }
        }
    } else {
        #pragma unroll
        for (int i = 0; i < STAGE_ITERS; ++i) {
            const int v = tid + i * THREADS;
            if (v < VEC_PER_BLOCK &&
                blockFloatBase + (long long)v * 4 + 4 <= totalFloats) {
                const unsigned lds_off = (unsigned)(v * 16);
                const uint64_t gaddr   = (uint64_t)(uintptr_t)(inBase + v * 4);
                asm volatile("global_load_async_to_lds_b128 %0, %1, off"
                             :: "v"(lds_off), "v"(gaddr)
                             : "memory");
            }
        }
    }
    // Wait for this wave's async transfers, then sync all waves of the block.
    asm volatile("s_wait_asynccnt 0x0" ::: "memory");
    __syncthreads();

    // ---------- Stage 2: one item per thread; LDS stride 34 dwords is
    // bank-conflict-free under wave32 (34*t mod 64 hits 32 distinct banks).
    if (blockItemBase + tid < (long long)n_items) {
        float* __restrict__ p = smem + tid * FLOATS_PER_ITEM;

        float xs[JOINTS], ys[JOINTS];
        float mnx = FLT_MAX,  mny = FLT_MAX;
        float mxx = -FLT_MAX, mxy = -FLT_MAX;
        bool  any = false;

        #pragma unroll
        for (int j = 0; j < JOINTS; ++j) {
            const float x = p[2 * j];
            const float y = p[2 * j + 1];
            xs[j] = x; ys[j] = y;
            const bool valid = (x != 0.0f) || (y != 0.0f);
            if (valid) {
                any = true;
                mnx = fminf(mnx, x); mxx = fmaxf(mxx, x);
                mny = fminf(mny, y); mxy = fmaxf(mxy, y);
            }
        }
        if (!any) { mnx = mny = mxx = mxy = 0.0f; }  // all-zero item: identity map

        float rx = mxx - mnx; rx = (rx == 0.0f) ? 1.0f : rx;
        float ry = mxy - mny; ry = (ry == 0.0f) ? 1.0f : ry;
        // One exact IEEE division per channel; per-element (x-mn)*s keeps the
        // cancellation-critical subtraction exact (<=1 ulp vs reference div).
        const float sx = 1.0f / rx;
        const float sy = 1.0f / ry;

        #pragma unroll
        for (int j = 0; j < JOINTS; ++j) {
            p[2 * j]     = (xs[j] - mnx) * sx;
            p[2 * j + 1] = (ys[j] - mny) * sy;
        }
    }
    __syncthreads();   // LDS results visible (dscnt drained) before async read-out

    // ---------- Stage 3: coalesced async copy LDS -> global (ASYNCcnt)
    if (fullBlock) {
        const uint64_t sbase = (uint64_t)(uintptr_t)outBase;
        #pragma unroll
        for (int i = 0; i < STAGE_ITERS; ++i) {
            const int v = tid + i * THREADS;
            if (i * THREADS + THREADS <= VEC_PER_BLOCK || v < VEC_PER_BLOCK) {
                const unsigned lds_off = (unsigned)(v * 16);
                const unsigned goff    = (unsigned)(v * 16);
                asm volatile("global_store_async_from_lds_b128 %0, %1, %2"
                             :: "v"(goff), "v"(lds_off), "s"(sbase)
                             : "memory");
            }
        }
    } else {
        #pragma unroll
        for (int i = 0; i < STAGE_ITERS; ++i) {
            const int v = tid + i * THREADS;
            if (v < VEC_PER_BLOCK &&
                blockFloatBase + (long long)v * 4 + 4 <= totalFloats) {
                const unsigned lds_off = (unsigned)(v * 16);
                const uint64_t gaddr   = (uint64_t)(uintptr_t)(outBase + v * 4);
                asm volatile("global_store_async_from_lds_b128 %0, %1, off"
                             :: "v"(gaddr), "v"(lds_off)
                             : "memory");
            }
        }
    }
    // Defensive drain (S_ENDPGM also implies S_WAIT_IDLE across all counters).
    asm volatile("s_wait_asynccnt 0x0" ::: "memory");
}

extern "C" void kernel_launch(void* const* d_in, const int* in_sizes, int n_in,
                              void* d_out, int out_size, void* d_ws, size_t ws_size,
                              hipStream_t stream)
{
    (void)n_in; (void)out_size; (void)d_ws; (void)ws_size;
    const float* in  = (const float*)d_in[0];
    float*       out = (float*)d_out;

    const long long totalFloats = (long long)in_sizes[0];
    const int n_items = (int)(totalFloats / FLOATS_PER_ITEM);   // 2,097,152
    const int blocks  = (n_items + ITEMS_PER_BLOCK - 1) / ITEMS_PER_BLOCK;
    const size_t smemBytes = (size_t)ITEMS_PER_BLOCK * FLOATS_PER_ITEM * sizeof(float);

    SkeletonPreprocessor_46797963657351_kernel<<<blocks, THREADS, smemBytes, stream>>>(
        in, out, n_items);
}